// pointnet2_5549097746749
// MI455X (gfx1250) — compile-verified
//
#include <hip/hip_runtime.h>

// CDNA5 / gfx1250, wave32.
typedef __attribute__((ext_vector_type(2))) float v2f;
typedef __attribute__((ext_vector_type(8))) float v8f;

#define BATCH 8
#define NPTS  8192
#define SPTS  2048
#define EPSV  1e-8f
#define FLT_BIG 3.0e38f

// ---------------------------------------------------------------------------
// Kernel 1: zero-fill the 512MB output at full HBM bandwidth (B128 stores).
// ---------------------------------------------------------------------------
__global__ void zero_fill_f4(float4* __restrict__ out, long long nvec) {
  long long i = (long long)blockIdx.x * blockDim.x + threadIdx.x;
  if (i < nvec) out[i] = make_float4(0.f, 0.f, 0.f, 0.f);
}

// Branchless min-3 insert using v_med3_f32 (no NaN canonicalization, 3
// independent ops per element). Keeps a0 <= a1 <= a2 = three smallest seen.
//   a0' = med3(x, a0, -BIG) == min(x, a0)
//   a1' = med3(x, a0, a1)   == new 2nd smallest
//   a2' = med3(x, a1, a2)   == new 3rd smallest
__device__ __forceinline__ void push3(float x, float& a0, float& a1, float& a2) {
  const float n0 = __builtin_amdgcn_fmed3f(x, a0, -FLT_BIG);
  const float n1 = __builtin_amdgcn_fmed3f(x, a0, a1);
  const float n2 = __builtin_amdgcn_fmed3f(x, a1, a2);
  a0 = n0; a1 = n1; a2 = n2;
}

// ---------------------------------------------------------------------------
// Kernel 2: per wave: 16 src rows vs all 2048 dst cols, two passes.
//   Pass 1: WMMA f32 16x16x4 tiles of (|d|^2 - 2 s.d); track 3 smallest
//           values per row (value-only, branchless), shfl-merge across lanes.
//   Pass 2: recompute tiles; any element <= third-smallest is a top-3 member,
//           store its normalized inverse-distance weight directly.
// ---------------------------------------------------------------------------
__global__ __launch_bounds__(256) void knn3_interp(const float* __restrict__ src,
                                                   const float* __restrict__ dst,
                                                   float* __restrict__ out) {
  __shared__ float4 dstS[SPTS];              // 32KB: (x, y, z, |d|^2)

  const int b    = blockIdx.y;
  const int tid  = threadIdx.x;
  const int wave = tid >> 5;
  const int lane = tid & 31;
  const int cl   = lane & 15;                // column lane within 16
  const bool hi  = lane >= 16;

  for (int i = tid; i < SPTS; i += 256) {
    const float x = dst[((size_t)b * SPTS + i) * 3 + 0];
    const float y = dst[((size_t)b * SPTS + i) * 3 + 1];
    const float z = dst[((size_t)b * SPTS + i) * 3 + 2];
    dstS[i] = make_float4(x, y, z, x * x + y * y + z * z);
  }
  __syncthreads();

  const int rowTile = (blockIdx.x * 8 + wave) * 16;   // 16 src rows per wave

  // A-matrix (16x4 f32): row m = [sx, sy, sz, 1].
  // ISA layout: lanes 0-15 -> {K=0,K=1}, lanes 16-31 -> {K=2,K=3}.
  const int arow = rowTile + cl;
  const float sx = src[((size_t)b * NPTS + arow) * 3 + 0];
  const float sy = src[((size_t)b * NPTS + arow) * 3 + 1];
  const float sz = src[((size_t)b * NPTS + arow) * 3 + 2];
  v2f a;
  a.x = hi ? sz   : sx;
  a.y = hi ? 1.0f : sy;

  // |s|^2 for the 8 rows this lane's C/D VGPRs map to
  // (C/D layout: VGPR r -> row r for lanes 0-15, row r+8 for lanes 16-31).
  float s2r[8];
#pragma unroll
  for (int r = 0; r < 8; ++r) {
    const int row = rowTile + r + (hi ? 8 : 0);
    const float qx = src[((size_t)b * NPTS + row) * 3 + 0];
    const float qy = src[((size_t)b * NPTS + row) * 3 + 1];
    const float qz = src[((size_t)b * NPTS + row) * 3 + 2];
    s2r[r] = qx * qx + qy * qy + qz * qz;
  }

  float a0[8], a1[8], a2[8];
#pragma unroll
  for (int r = 0; r < 8; ++r) { a0[r] = a1[r] = a2[r] = FLT_BIG; }

  // ---- Pass 1: scan all columns, keep 3 smallest partial distances ----
#pragma unroll 2
  for (int t = 0; t < SPTS / 16; ++t) {
    const float4 p = dstS[t * 16 + cl];
    // B-matrix (4x16): col n = [-2dx, -2dy, -2dz, |d|^2] (layout mirrors A).
    v2f bm;
    bm.x = hi ? (-2.0f * p.z) : (-2.0f * p.x);
    bm.y = hi ? (p.w)         : (-2.0f * p.y);
    v8f acc = {};
    acc = __builtin_amdgcn_wmma_f32_16x16x4_f32(
        false, a, false, bm, (short)0, acc, false, false);
#pragma unroll
    for (int r = 0; r < 8; ++r) push3(acc[r], a0[r], a1[r], a2[r]);
  }

  // Butterfly merge across the 16 column lanes of each half (steps 1,2,4,8;
  // rows 0-7 live in lanes 0-15, rows 8-15 in lanes 16-31).
#pragma unroll
  for (int step = 1; step < 16; step <<= 1) {
#pragma unroll
    for (int r = 0; r < 8; ++r) {
      const float p0 = __shfl_xor(a0[r], step, 32);
      const float p1 = __shfl_xor(a1[r], step, 32);
      const float p2 = __shfl_xor(a2[r], step, 32);
      push3(p0, a0[r], a1[r], a2[r]);
      push3(p1, a0[r], a1[r], a2[r]);
      push3(p2, a0[r], a1[r], a2[r]);
    }
  }

  // Per-row threshold (3rd smallest partial) and normalization factor.
  float thr[8], winv[8];
#pragma unroll
  for (int r = 0; r < 8; ++r) {
    const float w0 = 1.0f / (a0[r] + s2r[r] + EPSV);
    const float w1 = 1.0f / (a1[r] + s2r[r] + EPSV);
    const float w2 = 1.0f / (a2[r] + s2r[r] + EPSV);
    winv[r] = 1.0f / (w0 + w1 + w2);
    thr[r]  = a2[r];
  }

  // ---- Pass 2: recompute tiles, store weights for top-3 members ----
#pragma unroll 2
  for (int t = 0; t < SPTS / 16; ++t) {
    const int col = t * 16 + cl;
    const float4 p = dstS[col];
    v2f bm;
    bm.x = hi ? (-2.0f * p.z) : (-2.0f * p.x);
    bm.y = hi ? (p.w)         : (-2.0f * p.y);
    v8f acc = {};
    acc = __builtin_amdgcn_wmma_f32_16x16x4_f32(
        false, a, false, bm, (short)0, acc, false, false);
#pragma unroll
    for (int r = 0; r < 8; ++r) {
      const float d = acc[r];
      if (d <= thr[r]) {
        const int row = rowTile + r + (hi ? 8 : 0);
        out[((size_t)b * NPTS + row) * SPTS + col] =
            (1.0f / (d + s2r[r] + EPSV)) * winv[r];
      }
    }
  }
}

// ---------------------------------------------------------------------------
extern "C" void kernel_launch(void* const* d_in, const int* in_sizes, int n_in,
                              void* d_out, int out_size, void* d_ws, size_t ws_size,
                              hipStream_t stream) {
  (void)in_sizes; (void)n_in; (void)d_ws; (void)ws_size;
  const float* src = (const float*)d_in[0];   // [8, 8192, 3]
  const float* dst = (const float*)d_in[1];   // [8, 2048, 3]
  float* out = (float*)d_out;                 // [8, 8192, 2048]

  const long long nvec = (long long)out_size / 4;       // float4 count
  const int zgrid = (int)((nvec + 255) / 256);
  zero_fill_f4<<<zgrid, 256, 0, stream>>>((float4*)out, nvec);

  dim3 grid(NPTS / 16 / 8, BATCH);                      // (64, 8), 8 waves/block
  knn3_interp<<<grid, 256, 0, stream>>>(src, dst, out);
}